// HGAttn_89730456748202
// MI455X (gfx1250) — compile-verified
//
#include <hip/hip_runtime.h>
#include <hip/hip_bf16.h>

#define N_NODES 50000
#define M_EDGES 20000
#define N_INC   800000
#define HID     32
#define SLOPE   0.2f
#define BN_EPS  1e-5f

typedef float v2f __attribute__((ext_vector_type(2)));
typedef float v8f __attribute__((ext_vector_type(8)));

__device__ __forceinline__ float leaky(float x) { return x > 0.f ? x : SLOPE * x; }

// monotonic unsigned key for float atomicMax
__device__ __forceinline__ unsigned fkey(float v) {
    unsigned b = __float_as_uint(v);
    return (b & 0x80000000u) ? ~b : (b | 0x80000000u);
}
__device__ __forceinline__ float fkey_inv(unsigned k) {
    unsigned b = (k & 0x80000000u) ? (k ^ 0x80000000u) : ~k;
    return __uint_as_float(b);
}

// ---------------- BatchNorm ----------------
__global__ void bn_stats_kernel(const float* __restrict__ x, float* __restrict__ stats) {
    __shared__ float ls[64];
    int t = threadIdx.x;
    if (t < 64) ls[t] = 0.f;
    __syncthreads();
    int c = t & 31;
    int rpb = blockDim.x >> 5;
    float s = 0.f, s2 = 0.f;
    for (int r = blockIdx.x * rpb + (t >> 5); r < N_NODES; r += gridDim.x * rpb) {
        float v = x[r * HID + c];
        s += v; s2 += v * v;
    }
    atomicAdd(&ls[c], s);
    atomicAdd(&ls[32 + c], s2);
    __syncthreads();
    if (t < 64) atomicAdd(&stats[t], ls[t]);
}

__global__ void bn_apply_kernel(const float* __restrict__ x, const float* __restrict__ stats,
                                const float* __restrict__ gamma, const float* __restrict__ beta,
                                float* __restrict__ y) {
    int i = blockIdx.x * blockDim.x + threadIdx.x;
    if (i >= N_NODES * HID) return;
    int c = i & 31;
    const float inv_n = 1.0f / (float)N_NODES;
    float mean = stats[c] * inv_n;
    float var  = stats[32 + c] * inv_n - mean * mean;
    y[i] = (x[i] - mean) * rsqrtf(var + BN_EPS) * gamma[c] + beta[c];
}

// ---------------- degrees D (nodes), B (edges) ----------------
__global__ void degrees_kernel(const int* __restrict__ ni, const int* __restrict__ ei,
                               float* __restrict__ D, float* __restrict__ B) {
    int i = blockIdx.x * blockDim.x + threadIdx.x;
    if (i >= N_INC) return;
    atomicAdd(&D[ni[i]], 1.f);
    atomicAdd(&B[ei[i]], 1.f);
}

// ---------------- segment-sum of node features by edge ----------------
__global__ void seg_sum_edge_kernel(const float* __restrict__ x, const int* __restrict__ ni,
                                    const int* __restrict__ ei, float* __restrict__ ef) {
    int i = blockIdx.x * blockDim.x + threadIdx.x;
    if (i >= N_INC * HID) return;
    int e = i >> 5, c = i & 31;
    atomicAdd(&ef[ei[e] * HID + c], x[ni[e] * HID + c]);
}

// ---------------- WMMA GEMM: C[M,N] = A[M,32] @ B[32,N], fp32, wave-per-tile ----------------
__global__ void gemm_k32_wmma(const float* __restrict__ A, const float* __restrict__ B,
                              float* __restrict__ C, int M, int N) {
    const int lane = threadIdx.x & 31;
    const int wave = threadIdx.x >> 5;
    const int wavesPerBlock = blockDim.x >> 5;
    const int ntiles = N >> 4;
    const int totalTiles = (M >> 4) * ntiles;
    int tile = blockIdx.x * wavesPerBlock + wave;
    if (tile >= totalTiles) return;   // wave-uniform: EXEC stays all-1s for active waves
    const int mt = tile / ntiles;
    const int nt = tile - mt * ntiles;
    const int half = lane >> 4;               // 0: K pair {0,1}, 1: K pair {2,3}
    const int mrow = (mt << 4) + (lane & 15); // A row for this lane
    const int ncol = (nt << 4) + (lane & 15); // B/C column for this lane
    v8f c = {0.f, 0.f, 0.f, 0.f, 0.f, 0.f, 0.f, 0.f};
    #pragma unroll
    for (int k0 = 0; k0 < 32; k0 += 4) {
        const int ka = k0 + (half << 1);
        v2f a, b;
        a.x = A[mrow * 32 + ka];
        a.y = A[mrow * 32 + ka + 1];
        b.x = B[ka * N + ncol];
        b.y = B[(ka + 1) * N + ncol];
        c = __builtin_amdgcn_wmma_f32_16x16x4_f32(false, a, false, b, (short)0, c, false, false);
    }
    #pragma unroll
    for (int g = 0; g < 8; ++g) {
        C[((mt << 4) + g + (half << 3)) * N + ncol] = c[g];
    }
}

// ---------------- attention score projection: s[r,h] = <row(r,h), att[h]> ----------------
__global__ void scores_kernel(const float* __restrict__ Xh, const float* __restrict__ att,
                              int rows, int H, int attOff, float* __restrict__ s) {
    int i = blockIdx.x * blockDim.x + threadIdx.x;
    if (i >= rows * H) return;
    int r = i / H, h = i - r * H;
    const float* xr = Xh + (r * H + h) * 32;
    const float* ar = att + h * 64 + attOff;   // att row stride = 2F = 64
    float acc = 0.f;
    #pragma unroll
    for (int f = 0; f < 32; ++f) acc += xr[f] * ar[f];
    s[i] = acc;
}

// ---------------- segmented softmax: max pass ----------------
__global__ void alpha_max_kernel(const float* __restrict__ sn, const float* __restrict__ se,
                                 const int* __restrict__ ni, const int* __restrict__ ei,
                                 unsigned* __restrict__ amax, int H) {
    int i = blockIdx.x * blockDim.x + threadIdx.x;
    if (i >= N_INC * H) return;
    int e = i / H, h = i - e * H;
    float a = leaky(sn[ni[e] * H + h] + se[ei[e] * H + h]);
    atomicMax(&amax[ei[e] * H + h], fkey(a));
}

// ---------------- segmented softmax: exp + denom pass ----------------
__global__ void alpha_exp_kernel(const float* __restrict__ sn, const float* __restrict__ se,
                                 const int* __restrict__ ni, const int* __restrict__ ei,
                                 const unsigned* __restrict__ amax, float* __restrict__ ea,
                                 float* __restrict__ denom, int H) {
    int i = blockIdx.x * blockDim.x + threadIdx.x;
    if (i >= N_INC * H) return;
    int e = i / H, h = i - e * H;
    float a = leaky(sn[ni[e] * H + h] + se[ei[e] * H + h]);
    float m = fkey_inv(amax[ei[e] * H + h]);
    float x = expf(a - m);
    ea[i] = x;
    atomicAdd(&denom[ei[e] * H + h], x);
}

// ---------------- propagate nodes -> hyperedges ----------------
__global__ void prop_to_edges_kernel(const float* __restrict__ Xh, const float* __restrict__ ea,
                                     const float* __restrict__ denom, const float* __restrict__ B,
                                     const int* __restrict__ ni, const int* __restrict__ ei,
                                     float* __restrict__ oute, int H) {
    int i = blockIdx.x * blockDim.x + threadIdx.x;
    if (i >= N_INC * H) return;
    int e = i / H, h = i - e * H;
    int n = ni[e], m = ei[e];
    float alpha = ea[i] / denom[m * H + h];
    float bm = B[m];
    float w = (bm > 0.f ? 1.f / bm : 0.f) * alpha;
    const float* xr = Xh + (n * H + h) * 32;
    float* orow = oute + (m * H + h) * 32;
    #pragma unroll
    for (int f = 0; f < 32; ++f) atomicAdd(&orow[f], w * xr[f]);
}

// ---------------- propagate hyperedges -> nodes ----------------
__global__ void prop_to_nodes_kernel(const float* __restrict__ oute, const float* __restrict__ ea,
                                     const float* __restrict__ denom, const float* __restrict__ D,
                                     const int* __restrict__ ni, const int* __restrict__ ei,
                                     float* __restrict__ nacc, int H) {
    int i = blockIdx.x * blockDim.x + threadIdx.x;
    if (i >= N_INC * H) return;
    int e = i / H, h = i - e * H;
    int n = ni[e], m = ei[e];
    float alpha = ea[i] / denom[m * H + h];
    float dn = D[n];
    float w = (dn > 0.f ? 1.f / dn : 0.f) * alpha;
    const float* er = oute + (m * H + h) * 32;
    float* nrow = nacc + (n * H + h) * 32;
    #pragma unroll
    for (int f = 0; f < 32; ++f) atomicAdd(&nrow[f], w * er[f]);
}

// ---------------- mean over heads + bias + leaky ----------------
__global__ void finalize_kernel(const float* __restrict__ nacc, const float* __restrict__ bias,
                                float* __restrict__ y, int H) {
    int i = blockIdx.x * blockDim.x + threadIdx.x;
    if (i >= N_NODES * HID) return;
    int r = i >> 5, c = i & 31;
    float s = 0.f;
    for (int h = 0; h < H; ++h) s += nacc[(r * H + h) * 32 + c];
    s = s * (1.f / (float)H) + bias[c];
    y[i] = leaky(s);
}

static inline int cdiv(int a, int b) { return (a + b - 1) / b; }

extern "C" void kernel_launch(void* const* d_in, const int* in_sizes, int n_in,
                              void* d_out, int out_size, void* d_ws, size_t ws_size,
                              hipStream_t stream) {
    const float* node_fea = (const float*)d_in[0];
    const int*   e        = (const int*)d_in[1];
    const float* bn_gamma = (const float*)d_in[2];
    const float* bn_beta  = (const float*)d_in[3];
    const float* W1       = (const float*)d_in[4];
    const float* att1     = (const float*)d_in[5];
    const float* b1       = (const float*)d_in[6];
    const float* W2       = (const float*)d_in[7];
    const float* att2     = (const float*)d_in[8];
    const float* b2       = (const float*)d_in[9];

    const int* ni = e;            // node_idx
    const int* ei = e + N_INC;    // edge_idx

    // workspace layout (floats)
    float* ws = (float*)d_ws;
    size_t o = 0;
    float*    stats = ws + o; o += 64;
    float*    xbn   = ws + o; o += (size_t)N_NODES * HID;
    float*    ef    = ws + o; o += (size_t)M_EDGES * HID;
    float*    Xh    = ws + o; o += (size_t)N_NODES * 128;
    float*    Eh    = ws + o; o += (size_t)M_EDGES * 128;
    float*    sn    = ws + o; o += (size_t)N_NODES * 4;
    float*    se    = ws + o; o += (size_t)M_EDGES * 4;
    float*    ea    = ws + o; o += (size_t)N_INC * 4;
    unsigned* amax  = (unsigned*)(ws + o); o += (size_t)M_EDGES * 4;
    float*    denom = ws + o; o += (size_t)M_EDGES * 4;
    float*    Dd    = ws + o; o += (size_t)N_NODES;
    float*    Bb    = ws + o; o += (size_t)M_EDGES;
    float*    oute  = ws + o; o += (size_t)M_EDGES * 128;
    float*    nacc  = ws + o; o += (size_t)N_NODES * 128;
    float*    x1    = ws + o; o += (size_t)N_NODES * HID;

    const int T = 256;

    // ---- BatchNorm ----
    hipMemsetAsync(stats, 0, 64 * sizeof(float), stream);
    bn_stats_kernel<<<512, 256, 0, stream>>>(node_fea, stats);
    bn_apply_kernel<<<cdiv(N_NODES * HID, T), T, 0, stream>>>(node_fea, stats, bn_gamma, bn_beta, xbn);

    // ---- degrees (shared by both layers) ----
    hipMemsetAsync(Dd, 0, N_NODES * sizeof(float), stream);
    hipMemsetAsync(Bb, 0, M_EDGES * sizeof(float), stream);
    degrees_kernel<<<cdiv(N_INC, T), T, 0, stream>>>(ni, ei, Dd, Bb);

    // ================= Layer 1 (H=4, N=128) =================
    hipMemsetAsync(ef, 0, (size_t)M_EDGES * HID * sizeof(float), stream);
    seg_sum_edge_kernel<<<cdiv(N_INC * HID, T), T, 0, stream>>>(xbn, ni, ei, ef);

    gemm_k32_wmma<<<cdiv((N_NODES / 16) * 8, 4), 128, 0, stream>>>(xbn, W1, Xh, N_NODES, 128);
    gemm_k32_wmma<<<cdiv((M_EDGES / 16) * 8, 4), 128, 0, stream>>>(ef,  W1, Eh, M_EDGES, 128);

    scores_kernel<<<cdiv(N_NODES * 4, T), T, 0, stream>>>(Xh, att1, N_NODES, 4, 0,  sn);
    scores_kernel<<<cdiv(M_EDGES * 4, T), T, 0, stream>>>(Eh, att1, M_EDGES, 4, 32, se);

    hipMemsetAsync(amax, 0, (size_t)M_EDGES * 4 * sizeof(unsigned), stream);
    alpha_max_kernel<<<cdiv(N_INC * 4, T), T, 0, stream>>>(sn, se, ni, ei, amax, 4);
    hipMemsetAsync(denom, 0, (size_t)M_EDGES * 4 * sizeof(float), stream);
    alpha_exp_kernel<<<cdiv(N_INC * 4, T), T, 0, stream>>>(sn, se, ni, ei, amax, ea, denom, 4);

    hipMemsetAsync(oute, 0, (size_t)M_EDGES * 128 * sizeof(float), stream);
    prop_to_edges_kernel<<<cdiv(N_INC * 4, T), T, 0, stream>>>(Xh, ea, denom, Bb, ni, ei, oute, 4);
    hipMemsetAsync(nacc, 0, (size_t)N_NODES * 128 * sizeof(float), stream);
    prop_to_nodes_kernel<<<cdiv(N_INC * 4, T), T, 0, stream>>>(oute, ea, denom, Dd, ni, ei, nacc, 4);

    finalize_kernel<<<cdiv(N_NODES * HID, T), T, 0, stream>>>(nacc, b1, x1, 4);

    // ================= Layer 2 (H=1, N=32) =================
    hipMemsetAsync(ef, 0, (size_t)M_EDGES * HID * sizeof(float), stream);
    seg_sum_edge_kernel<<<cdiv(N_INC * HID, T), T, 0, stream>>>(x1, ni, ei, ef);

    gemm_k32_wmma<<<cdiv((N_NODES / 16) * 2, 4), 128, 0, stream>>>(x1, W2, Xh, N_NODES, 32);
    gemm_k32_wmma<<<cdiv((M_EDGES / 16) * 2, 4), 128, 0, stream>>>(ef, W2, Eh, M_EDGES, 32);

    scores_kernel<<<cdiv(N_NODES, T), T, 0, stream>>>(Xh, att2, N_NODES, 1, 0,  sn);
    scores_kernel<<<cdiv(M_EDGES, T), T, 0, stream>>>(Eh, att2, M_EDGES, 1, 32, se);

    hipMemsetAsync(amax, 0, (size_t)M_EDGES * sizeof(unsigned), stream);
    alpha_max_kernel<<<cdiv(N_INC, T), T, 0, stream>>>(sn, se, ni, ei, amax, 1);
    hipMemsetAsync(denom, 0, (size_t)M_EDGES * sizeof(float), stream);
    alpha_exp_kernel<<<cdiv(N_INC, T), T, 0, stream>>>(sn, se, ni, ei, amax, ea, denom, 1);

    hipMemsetAsync(oute, 0, (size_t)M_EDGES * HID * sizeof(float), stream);
    prop_to_edges_kernel<<<cdiv(N_INC, T), T, 0, stream>>>(Xh, ea, denom, Bb, ni, ei, oute, 1);
    hipMemsetAsync(nacc, 0, (size_t)N_NODES * HID * sizeof(float), stream);
    prop_to_nodes_kernel<<<cdiv(N_INC, T), T, 0, stream>>>(oute, ea, denom, Dd, ni, ei, nacc, 1);

    finalize_kernel<<<cdiv(N_NODES * HID, T), T, 0, stream>>>(nacc, b2, (float*)d_out, 1);
}